// PICNNEntropicOTQuantileRegression_62294205662143
// MI455X (gfx1250) — compile-verified
//
#include <hip/hip_runtime.h>

// ---------------------------------------------------------------------------
// PICNN entropic-OT dual over all (i,j) pairs, n=768, xd=16, yd=8, ud=zd=64.
// Compute-bound (~6.5 GFLOP, ~150KB data). eps=0.01 amplifies psi error x100
// in the exponent -> use full-f32 WMMA (V_WMMA_F32_16X16X4_F32), not f16.
// Gates fold into column-scaled A matrices (per-block, in LDS); biases fold
// into an augmented K column. One block per i, 4 waves, each wave owns whole
// j-tiles (16 columns) -> no barriers in the hot loop.
// ---------------------------------------------------------------------------

typedef float v2f __attribute__((ext_vector_type(2)));
typedef float v8f __attribute__((ext_vector_type(8)));

#define EPS      0.01f
#define INV_EPS  100.0f
#define LOG_N    6.6437898f   // ln(768)
#define A1_LD    76           // 64 (z) + 8 (y) + 1 (bias) + 3 pad
#define A0_LD    12           // 8 (y) + 1 (bias) + 3 pad
#define STG_LD   18           // 16 + 2 pad: lo/hi lane halves hit disjoint banks

__device__ __forceinline__ v8f wmma_f32(v2f a, v2f b, v8f c) {
  // D(16x16,f32) = A(16x4,f32) * B(4x16,f32) + C   -- v_wmma_f32_16x16x4_f32
  return __builtin_amdgcn_wmma_f32_16x16x4_f32(false, a, false, b,
                                               (short)0, c, false, false);
}

struct PicnnArgs {
  const float *X, *U, *Y;
  const float *Wu0, *bu0, *Wyu0, *byu0, *Wy0, *Wx0, *b0;
  const float *Wzu1, *bzu1, *Wz1, *Wyu1, *byu1, *Wy1, *Wx1, *b1;
  const float *Wzu2, *bzu2, *Wz2, *Wyu2, *byu2, *Wy2, *Wx2, *b2;
  float *out;
  int n;
};

__global__ __launch_bounds__(128) void picnn_entropic_lse(PicnnArgs P) {
  const int i    = blockIdx.x;
  const int t    = threadIdx.x;
  const int lane = t & 31;
  const int wave = t >> 5;
  const int hi   = lane >> 4;   // 0: lanes 0-15, 1: lanes 16-31
  const int nn   = lane & 15;   // column (N) / row-within-tile (M for A)

  __shared__ float A1ext[64 * A1_LD];        // [Wz1*g1 | Wy1*gy1 | c1 | 0]
  __shared__ float A0ext[64 * A0_LD];        // [Wy0*gy0 | c0 | 0]
  __shared__ float stage[4][72 * STG_LD];    // per-wave B1 staging: z0(64)+y(8)
  __shared__ float ush[64], g1sh[64], c0sh[64], c1sh[64], wz2gsh[64];
  __shared__ float gy0sh[8], gy1sh[8], vysh[8];
  __shared__ float xsh[16];
  __shared__ float c2sh;
  __shared__ float redm[4], reds[4];

  // ---- phase 0: stage x_i -------------------------------------------------
  if (t < 16) xsh[t] = P.X[(size_t)i * 16 + t];
  __syncthreads();

  // ---- phase 1: u = relu(Wu0 x + bu0) ------------------------------------
  if (t < 64) {
    float a = P.bu0[t];
#pragma unroll
    for (int k = 0; k < 16; ++k) a += P.Wu0[t * 16 + k] * xsh[k];
    ush[t] = fmaxf(a, 0.f);
  }
  __syncthreads();

  // ---- phase 2: all per-i gates / biases ----------------------------------
  if (t < 64) {
    float a1 = P.bzu1[t], a2 = P.b1[t], a3 = P.bzu2[t];
    for (int k = 0; k < 64; ++k) {
      const float uk = ush[k];
      a1 += P.Wzu1[t * 64 + k] * uk;   // gate_z1 pre
      a2 += P.Wx1 [t * 64 + k] * uk;   // c1
      a3 += P.Wzu2[t * 64 + k] * uk;   // gate_z2 pre
    }
    g1sh[t]   = fmaxf(a1, 0.f);
    c1sh[t]   = a2;
    wz2gsh[t] = P.Wz2[t] * fmaxf(a3, 0.f);   // Wz2 (1,64) elementwise-gated
    float a0 = P.b0[t];
#pragma unroll
    for (int k = 0; k < 16; ++k) a0 += P.Wx0[t * 16 + k] * xsh[k];
    c0sh[t] = a0;                             // c0 = Wx0 x + b0
  } else if (t < 72) {
    const int o = t - 64;
    float a0 = P.byu0[o];
#pragma unroll
    for (int k = 0; k < 16; ++k) a0 += P.Wyu0[o * 16 + k] * xsh[k];
    gy0sh[o] = a0;
    float a1 = P.byu1[o], a2 = P.byu2[o];
    for (int k = 0; k < 64; ++k) {
      const float uk = ush[k];
      a1 += P.Wyu1[o * 64 + k] * uk;
      a2 += P.Wyu2[o * 64 + k] * uk;
    }
    gy1sh[o] = a1;
    vysh[o]  = P.U[(size_t)i * 8 + o] - P.Wy2[o] * a2;  // (U_i - Wy2*gy2)
  } else if (t == 72) {
    float a = P.b2[0];
    for (int k = 0; k < 64; ++k) a += P.Wx2[k] * ush[k];
    c2sh = a;                                 // c2 = Wx2 u + b2
  }
  __syncthreads();

  // ---- phase 3: build column-scaled, bias-augmented A matrices ------------
  if (t < 64) {
    const int m = t;
#pragma unroll
    for (int k = 0; k < 8; ++k) A0ext[m * A0_LD + k] = P.Wy0[m * 8 + k] * gy0sh[k];
    A0ext[m * A0_LD + 8]  = c0sh[m];
    A0ext[m * A0_LD + 9]  = 0.f;
    A0ext[m * A0_LD + 10] = 0.f;
    A0ext[m * A0_LD + 11] = 0.f;
    for (int k = 0; k < 64; ++k) A1ext[m * A1_LD + k] = P.Wz1[m * 64 + k] * g1sh[k];
#pragma unroll
    for (int k = 0; k < 8; ++k) A1ext[m * A1_LD + 64 + k] = P.Wy1[m * 8 + k] * gy1sh[k];
    A1ext[m * A1_LD + 72] = c1sh[m];
    A1ext[m * A1_LD + 73] = 0.f;
    A1ext[m * A1_LD + 74] = 0.f;
    A1ext[m * A1_LD + 75] = 0.f;
  }
  __syncthreads();

  // ---- persistent per-lane registers --------------------------------------
  // A0 fragments (4 M-tiles x 3 K-steps), ISA A-layout: v0=K even, v1=K odd,
  // lanes 16-31 shifted by K+2 -> contiguous float2 in row-major LDS.
  v2f a0frag[4][3];
#pragma unroll
  for (int mt = 0; mt < 4; ++mt)
#pragma unroll
    for (int ks = 0; ks < 3; ++ks)
      a0frag[mt][ks] = *(const v2f*)&A0ext[(mt * 16 + nn) * A0_LD + ks * 4 + 2 * hi];

  float wf[32];                              // wz2g in C/D fragment order
#pragma unroll
  for (int mt = 0; mt < 4; ++mt)
#pragma unroll
    for (int r = 0; r < 8; ++r)
      wf[mt * 8 + r] = wz2gsh[mt * 16 + r + 8 * hi];

  const float vy0 = vysh[2 * hi + 0], vy1 = vysh[2 * hi + 1];
  const float vy2 = vysh[2 * hi + 4], vy3 = vysh[2 * hi + 5];
  const float c2  = c2sh;

  float* stg = &stage[wave][0];
  float mrun = -3.0e38f, srun = 0.f;         // online logsumexp state
  const int ntiles = P.n >> 4;

  // ---- hot loop: each wave owns whole 16-column j-tiles -------------------
#pragma unroll 1
  for (int tj = wave; tj < ntiles; tj += 4) {
    const int j0 = tj * 16;
    const float* yrow = P.Y + (size_t)(j0 + nn) * 8;
    const v2f yA = *(const v2f*)(yrow + 2 * hi);       // y[2hi], y[2hi+1]
    const v2f yB = *(const v2f*)(yrow + 4 + 2 * hi);   // y[4+2hi], y[5+2hi]

    // stage raw y into B1 rows 64..71 (gating lives in A1)
    stg[(64 + 2 * hi) * STG_LD + nn] = yA.x;
    stg[(65 + 2 * hi) * STG_LD + nn] = yA.y;
    stg[(68 + 2 * hi) * STG_LD + nn] = yB.x;
    stg[(69 + 2 * hi) * STG_LD + nn] = yB.y;

    v2f bbias; bbias.x = hi ? 0.f : 1.f; bbias.y = 0.f; // rows [1,0,0,0]

    // GEMM0: z0_pre(64x16) = A0ext(64x12) @ [y;1;0](12x16), B from registers
#pragma unroll
    for (int mt = 0; mt < 4; ++mt) {
      v8f acc = {};
      acc = wmma_f32(a0frag[mt][0], yA, acc);
      acc = wmma_f32(a0frag[mt][1], yB, acc);
      acc = wmma_f32(a0frag[mt][2], bbias, acc);
#pragma unroll
      for (int r = 0; r < 8; ++r)            // relu -> staging rows 0..63
        stg[(mt * 16 + r + 8 * hi) * STG_LD + nn] = fmaxf(acc[r], 0.f);
    }

    // B1 fragments (same-wave LDS: hardware keeps DS in order, no barrier)
    v2f bf[19];
#pragma unroll
    for (int ks = 0; ks < 18; ++ks) {
      const int row = ks * 4 + 2 * hi;
      bf[ks].x = stg[row * STG_LD + nn];
      bf[ks].y = stg[(row + 1) * STG_LD + nn];
    }
    bf[18] = bbias;                          // bias rows 72..75 = [1,0,0,0]

    // GEMM1: z1_pre(64x16) = A1ext(64x76) @ [z0;y;1](76x16); fused head dot
    float dotz = 0.f;
#pragma unroll
    for (int mt = 0; mt < 4; ++mt) {
      v8f acc = {};
#pragma unroll
      for (int ks = 0; ks < 19; ++ks) {
        const v2f a = *(const v2f*)&A1ext[(mt * 16 + nn) * A1_LD + ks * 4 + 2 * hi];
        acc = wmma_f32(a, bf[ks], acc);
      }
#pragma unroll
      for (int r = 0; r < 8; ++r)
        dotz += wf[mt * 8 + r] * fmaxf(acc[r], 0.f);   // wz2g . relu(z1_pre)
    }
    dotz += __shfl_xor(dotz, 16, 32);        // combine lane halves per column
    float ydot = vy0 * yA.x + vy1 * yA.y + vy2 * yB.x + vy3 * yB.y;
    ydot += __shfl_xor(ydot, 16, 32);

    if (hi == 0) {                           // lanes 0-15 own the 16 columns
      const float slack = ydot - dotz - c2;  // U.y - psi
      const float nm = fmaxf(mrun, slack);
      srun = srun * __expf((mrun - nm) * INV_EPS) + __expf((slack - nm) * INV_EPS);
      mrun = nm;
    }
  }

  // ---- wave-level logsumexp butterfly (hi lanes carry identity -3e38/0) ---
#pragma unroll
  for (int off = 16; off >= 1; off >>= 1) {
    const float om = __shfl_xor(mrun, off, 32);
    const float os = __shfl_xor(srun, off, 32);
    const float nm = fmaxf(mrun, om);
    srun = srun * __expf((mrun - nm) * INV_EPS) + os * __expf((om - nm) * INV_EPS);
    mrun = nm;
  }
  if (lane == 0) { redm[wave] = mrun; reds[wave] = srun; }
  __syncthreads();
  if (t == 0) {
    float M = redm[0], S = reds[0];
#pragma unroll
    for (int w = 1; w < 4; ++w) {
      const float nm = fmaxf(M, redm[w]);
      S = S * __expf((M - nm) * INV_EPS) + reds[w] * __expf((redm[w] - nm) * INV_EPS);
      M = nm;
    }
    P.out[i] = EPS * (__logf(S) - LOG_N) + M;
  }
}

extern "C" void kernel_launch(void* const* d_in, const int* in_sizes, int n_in,
                              void* d_out, int out_size, void* d_ws, size_t ws_size,
                              hipStream_t stream) {
  (void)n_in; (void)out_size; (void)d_ws; (void)ws_size;
  PicnnArgs P;
  // setup_inputs() dict order: X, U, Y, then params in insertion order.
  P.X    = (const float*)d_in[0];
  P.U    = (const float*)d_in[1];
  P.Y    = (const float*)d_in[2];
  P.Wu0  = (const float*)d_in[3];  P.bu0  = (const float*)d_in[4];
  P.Wyu0 = (const float*)d_in[5];  P.byu0 = (const float*)d_in[6];
  P.Wy0  = (const float*)d_in[7];  P.Wx0  = (const float*)d_in[8];
  P.b0   = (const float*)d_in[9];
  P.Wzu1 = (const float*)d_in[10]; P.bzu1 = (const float*)d_in[11];
  P.Wz1  = (const float*)d_in[12]; P.Wyu1 = (const float*)d_in[13];
  P.byu1 = (const float*)d_in[14]; P.Wy1  = (const float*)d_in[15];
  P.Wx1  = (const float*)d_in[16]; P.b1   = (const float*)d_in[17];
  P.Wzu2 = (const float*)d_in[18]; P.bzu2 = (const float*)d_in[19];
  P.Wz2  = (const float*)d_in[20]; P.Wyu2 = (const float*)d_in[21];
  P.byu2 = (const float*)d_in[22]; P.Wy2  = (const float*)d_in[23];
  P.Wx2  = (const float*)d_in[24]; P.b2   = (const float*)d_in[25];
  P.out  = (float*)d_out;
  P.n    = in_sizes[0] / 16;       // X is [n,16] -> n = 768
  picnn_entropic_lse<<<dim3(P.n), dim3(128), 0, stream>>>(P);
}